// GraphSAGE_23940147708319
// MI455X (gfx1250) — compile-verified
//
#include <hip/hip_runtime.h>
#include <hip/hip_bf16.h>

typedef float v2f __attribute__((ext_vector_type(2)));
typedef float v8f __attribute__((ext_vector_type(8)));

// ---------------------------------------------------------------------------
// Degree count: one thread per edge, atomic add into cnt[dst].
// ---------------------------------------------------------------------------
__global__ void sage_degree(const int* __restrict__ dstIdx,
                            float* __restrict__ cnt, int E) {
    int e = blockIdx.x * blockDim.x + threadIdx.x;
    if (e < E) atomicAdd(&cnt[dstIdx[e]], 1.0f);
}

// ---------------------------------------------------------------------------
// Edge scatter: one wave (32 lanes) per edge; lane handles 4 floats per step.
// agg[dst][:] += h[src][:]   (L2-resident float atomics)
// ---------------------------------------------------------------------------
__global__ void sage_scatter(const float* __restrict__ Hin,
                             const int* __restrict__ srcIdx,
                             const int* __restrict__ dstIdx,
                             float* __restrict__ Agg, int E, int d) {
    int e = blockIdx.x * blockDim.y + threadIdx.y;
    if (e >= E) return;
    int s = srcIdx[e];
    int t = dstIdx[e];
    int lane = threadIdx.x;
    const float4* __restrict__ src = (const float4*)(Hin + (size_t)s * d);
    float* dst = Agg + (size_t)t * d;
    for (int f = lane; f * 4 < d; f += 32) {
        float4 v = src[f];
        atomicAdd(&dst[f * 4 + 0], v.x);
        atomicAdd(&dst[f * 4 + 1], v.y);
        atomicAdd(&dst[f * 4 + 2], v.z);
        atomicAdd(&dst[f * 4 + 3], v.w);
    }
}

// ---------------------------------------------------------------------------
// Fused SAGE linear:  Hout = act( (Agg/max(cnt,1)) @ Wl + Hin @ Wr + b )
//
// Block = 8 waves. Block covers 8 row tiles x (16*CB) output columns.
// Weights for the column group are staged cooperatively into LDS in a
// k-pair-interleaved layout [K/2][16*CB][2] so each lane's B fragment
// (W[ka][col], W[ka+1][col]) is a single aligned ds_load_b64.
//
// Per k-step each wave issues 2*CB V_WMMA_F32_16X16X4_F32, amortizing the
// A-fragment (node feature) loads across CB column tiles.
//
// Fragment layouts (ISA 7.12.2, f32):
//   A 16x4: lane l (lo=l&15, hi=l>>4): v0=A[lo][k+2hi], v1=A[lo][k+2hi+1]
//   B 4x16: mirrored: v0=B[k+2hi][col], v1=B[k+2hi+1][col]
//   C/D 16x16: VGPR g: lanes 0-15 -> row g, lanes 16-31 -> row g+8; col=lo
// ---------------------------------------------------------------------------
template <int CB>
__global__ void sage_gemm(const float* __restrict__ Hin,
                          const float* __restrict__ Agg,
                          const float* __restrict__ Cnt,
                          const float* __restrict__ Wl,
                          const float* __restrict__ Wr,
                          const float* __restrict__ Bias,
                          float* __restrict__ Hout,
                          int nRowTiles, int N, int K, int Nout, int doRelu) {
    const int W16 = 16 * CB;
    __shared__ float smem[16384];            // 64 KB: wls | wrs
    float* wls = smem;                       // K/2 x W16 x 2 floats
    float* wrs = smem + (size_t)K * W16;

    int tid = threadIdx.y * 32 + threadIdx.x;
    int colBase = blockIdx.y * W16;

    // Cooperative weight staging: coalesced global reads, interleaved LDS.
    for (int idx = tid; idx < K * W16; idx += 256) {
        int kk  = idx / W16;
        int col = idx - kk * W16;
        int colg = colBase + col;
        if (colg >= Nout) colg = Nout - 1;   // clamp (garbage cols never stored)
        size_t g = (size_t)kk * Nout + colg;
        int s = (kk >> 1) * (W16 * 2) + col * 2 + (kk & 1);
        wls[s] = Wl[g];
        wrs[s] = Wr[g];
    }
    __syncthreads();

    int rowTile = blockIdx.x * 8 + threadIdx.y;
    if (rowTile >= nRowTiles) return;        // wave-uniform exit

    int lane = threadIdx.x;
    int lo = lane & 15;
    int hi = lane >> 4;

    int rowA  = rowTile * 16 + lo;
    int rowAc = rowA < N ? rowA : N - 1;     // clamp loads in-bounds
    float invc = 1.0f / fmaxf(Cnt[rowAc], 1.0f);

    const float* __restrict__ aggRow = Agg + (size_t)rowAc * K;
    const float* __restrict__ hRow   = Hin + (size_t)rowAc * K;

    v8f acc[CB] = {};
    for (int k = 0; k < K; k += 4) {
        int ka = k + 2 * hi;
        v2f am = *(const v2f*)(aggRow + ka);
        am.x *= invc;
        am.y *= invc;
        v2f ah = *(const v2f*)(hRow + ka);
        int soff = (ka >> 1) * (W16 * 2) + lo * 2;
        #pragma unroll
        for (int c = 0; c < CB; ++c) {
            v2f bl = *(const v2f*)(wls + soff + c * 32);   // ds_load_b64
            v2f br = *(const v2f*)(wrs + soff + c * 32);   // ds_load_b64
            acc[c] = __builtin_amdgcn_wmma_f32_16x16x4_f32(
                         false, am, false, bl, (short)0, acc[c], false, false);
            acc[c] = __builtin_amdgcn_wmma_f32_16x16x4_f32(
                         false, ah, false, br, (short)0, acc[c], false, false);
        }
    }

    int rbase = rowTile * 16 + hi * 8;
    #pragma unroll
    for (int c = 0; c < CB; ++c) {
        int colB = colBase + c * 16 + lo;
        if (colB < Nout) {
            float b = Bias[colB];
            #pragma unroll
            for (int g = 0; g < 8; ++g) {
                int r = rbase + g;
                if (r < N) {
                    float v = acc[c][g] + b;
                    if (doRelu) v = fmaxf(v, 0.0f);
                    Hout[(size_t)r * Nout + colB] = v;
                }
            }
        }
    }
}

// ---------------------------------------------------------------------------
// Global mean pool over nodes + log_softmax. out[0..9]=pooled, out[10..19]=lsm.
// ---------------------------------------------------------------------------
__global__ void sage_pool(const float* __restrict__ H, float* __restrict__ out,
                          int n) {
    __shared__ float sums[256 * 10];
    int tid = threadIdx.x;
    float loc[10];
    #pragma unroll
    for (int j = 0; j < 10; ++j) loc[j] = 0.0f;
    for (int i = tid; i < n; i += 256) {
        const float* row = H + (size_t)i * 10;
        #pragma unroll
        for (int j = 0; j < 10; ++j) loc[j] += row[j];
    }
    #pragma unroll
    for (int j = 0; j < 10; ++j) sums[tid * 10 + j] = loc[j];
    __syncthreads();
    for (int s = 128; s > 0; s >>= 1) {
        if (tid < s) {
            #pragma unroll
            for (int j = 0; j < 10; ++j)
                sums[tid * 10 + j] += sums[(tid + s) * 10 + j];
        }
        __syncthreads();
    }
    if (tid == 0) {
        float p[10], m = -3.4e38f, lse = 0.0f;
        #pragma unroll
        for (int j = 0; j < 10; ++j) {
            p[j] = sums[j] / (float)n;
            m = fmaxf(m, p[j]);
        }
        #pragma unroll
        for (int j = 0; j < 10; ++j) lse += __expf(p[j] - m);
        lse = logf(lse) + m;
        #pragma unroll
        for (int j = 0; j < 10; ++j) {
            out[j]      = p[j];
            out[10 + j] = p[j] - lse;
        }
    }
}

// ---------------------------------------------------------------------------
// Launch: 6 SAGE layers (3-5 share w3) + pool. All async on `stream`.
// ---------------------------------------------------------------------------
extern "C" void kernel_launch(void* const* d_in, const int* in_sizes, int n_in,
                              void* d_out, int out_size, void* d_ws, size_t ws_size,
                              hipStream_t stream) {
    const float* x   = (const float*)d_in[0];
    const int*   ei  = (const int*)d_in[1];     // [2,E]: row0=src, row1=dst
    const float* w1l = (const float*)d_in[2];
    const float* w1r = (const float*)d_in[3];
    const float* b1  = (const float*)d_in[4];
    const float* w2l = (const float*)d_in[5];
    const float* w2r = (const float*)d_in[6];
    const float* b2  = (const float*)d_in[7];
    const float* w3l = (const float*)d_in[8];
    const float* w3r = (const float*)d_in[9];
    const float* b3  = (const float*)d_in[10];
    const float* w4l = (const float*)d_in[11];
    const float* w4r = (const float*)d_in[12];
    const float* b4  = (const float*)d_in[13];

    const int FIN = 128;
    int N = in_sizes[0] / FIN;
    int E = in_sizes[1] / 2;
    const int* srcIdx = ei;
    const int* dstIdx = ei + E;

    // Workspace partition (floats): cnt[N] | agg[N*256] | hA[N*256] | hB[N*256]
    float* cnt = (float*)d_ws;
    size_t cntPad = ((size_t)N + 255) & ~(size_t)255;
    float* agg = cnt + cntPad;
    float* hA  = agg + (size_t)N * 256;
    float* hB  = hA  + (size_t)N * 256;

    // Degrees (recomputed each call; no state carried across calls).
    hipMemsetAsync(cnt, 0, (size_t)N * sizeof(float), stream);
    sage_degree<<<(E + 255) / 256, 256, 0, stream>>>(dstIdx, cnt, E);

    auto run_layer = [&](const float* Hin, int K, int Nout, const float* Wl,
                         const float* Wr, const float* B, float* Hout, int relu) {
        hipMemsetAsync(agg, 0, (size_t)N * K * sizeof(float), stream);
        dim3 sblk(32, 8);
        sage_scatter<<<(E + 7) / 8, sblk, 0, stream>>>(Hin, srcIdx, dstIdx, agg, E, K);

        int rt = (N + 15) / 16;
        int rowBlocks = (rt + 7) / 8;
        // CB chosen so staged weights (2*K*16*CB floats) fit 64 KB LDS.
        int CB = (Nout % 64 == 0 && K <= 128) ? 4 : ((Nout % 32 == 0) ? 2 : 1);
        int colGroups = (Nout + 16 * CB - 1) / (16 * CB);
        dim3 grid(rowBlocks, colGroups);
        dim3 gblk(32, 8);
        if (CB == 4) {
            sage_gemm<4><<<grid, gblk, 0, stream>>>(Hin, agg, cnt, Wl, Wr, B,
                                                    Hout, rt, N, K, Nout, relu);
        } else if (CB == 2) {
            sage_gemm<2><<<grid, gblk, 0, stream>>>(Hin, agg, cnt, Wl, Wr, B,
                                                    Hout, rt, N, K, Nout, relu);
        } else {
            sage_gemm<1><<<grid, gblk, 0, stream>>>(Hin, agg, cnt, Wl, Wr, B,
                                                    Hout, rt, N, K, Nout, relu);
        }
    };

    run_layer(x,  128, 256, w1l, w1r, b1, hA, 1);   // sage1
    run_layer(hA, 256, 128, w2l, w2r, b2, hB, 1);   // sage2
    run_layer(hB, 128, 128, w3l, w3r, b3, hA, 1);   // sage3 (shared w)
    run_layer(hA, 128, 128, w3l, w3r, b3, hB, 1);   // sage3 again
    run_layer(hB, 128, 128, w3l, w3r, b3, hA, 1);   // sage3 again
    run_layer(hA, 128, 10,  w4l, w4r, b4, hB, 0);   // sage4 (no relu)

    sage_pool<<<1, 256, 0, stream>>>(hB, (float*)d_out, N);
}